// CPIGating_89644557402443
// MI455X (gfx1250) — compile-verified
//
#include <hip/hip_runtime.h>
#include <hip/hip_bf16.h>
#include <math.h>

// Problem constants (match reference)
#define BB  2
#define NN  2048
#define DD  256
#define HH  8
#define HD  32
#define MROWS (BB*NN)          // 4096 flattened rows
// 1/(sqrt(32)+1e-6)
#define INV_SCALE 0.17677666f

// LDS layout for the attention kernel (float indices, dynamic LDS only):
//   K tiles : 2 bufs x 16 rows x 36 floats  -> [0,    1152)
//   V tiles : 2 bufs x 16 rows x 36 floats  -> [1152, 2304)
//   P bufs  : 4 waves x 16 rows x 20 floats -> [2304, 3584)
#define KT_OFF   0
#define VT_OFF   1152
#define PB_OFF   2304
#define TILE_STRIDE 36         // 16B-aligned rows (36*4=144), conflict-free reads
#define SMEM_FLOATS 3584
#define SMEM_BYTES  (SMEM_FLOATS * 4)

typedef float v2f __attribute__((ext_vector_type(2)));
typedef float v8f __attribute__((ext_vector_type(8)));

// D = A(16x4 f32) * B(4x16 f32) + C, one wave, fp32 WMMA
__device__ __forceinline__ v8f wmma4(v2f a, v2f b, v8f c) {
  return __builtin_amdgcn_wmma_f32_16x16x4_f32(
      /*neg_a=*/false, a, /*neg_b=*/false, b,
      /*c_mod=*/(short)0, c, /*reuse_a=*/false, /*reuse_b=*/false);
}

__device__ __forceinline__ v2f ld2(const float* p) {
  return *reinterpret_cast<const v2f*>(p);
}

__device__ __forceinline__ float sigmoidf_(float x) {
  return 1.0f / (1.0f + __expf(-x));
}

// gfx1250 async copy: copies 16 bytes from global address gaddr to LDS byte offset lds_off, tracked by ASYNCcnt
__device__ __forceinline__ void async_copy_b128(unsigned lds_off,
                                                const float* gaddr) {
  asm volatile("global_load_async_to_lds_b128 %0, %1, off"
               :: "v"(lds_off), "v"(gaddr) : "memory");
}

// ---------------------------------------------------------------------------
// Kernel 1: q/k/v = x @ W^T + b.  grid=(MROWS/16, 3), block=128 (4 waves).
// Each wave: one 16-row x 64-col tile; A fragment reused across 4 B tiles.
// A 16x4 f32 frag: lane<16 -> (x[m][k],x[m][k+1]); lane>=16 -> (k+2,k+3).
// B 4x16 f32 frag: lane holds col n: (W[n][k+2*ksel], W[n][k+2*ksel+1]).
// ---------------------------------------------------------------------------
__global__ __launch_bounds__(128)
void qkv_proj_kernel(const float* __restrict__ x,
                     const float* __restrict__ wq, const float* __restrict__ bq,
                     const float* __restrict__ wk, const float* __restrict__ bk,
                     const float* __restrict__ wv, const float* __restrict__ bv,
                     float* __restrict__ qb, float* __restrict__ kb,
                     float* __restrict__ vb)
{
  const float* w;  const float* bias;  float* out;
  if (blockIdx.y == 0)      { w = wq; bias = bq; out = qb; }
  else if (blockIdx.y == 1) { w = wk; bias = bk; out = kb; }
  else                      { w = wv; bias = bv; out = vb; }

  const int wave = threadIdx.x >> 5;
  const int lane = threadIdx.x & 31;
  const int lh   = lane & 15;
  const int ksel = lane >> 4;          // 0 or 1 (K half of fragment)

  const int mt = blockIdx.x;           // 16-row tile
  const int m  = mt * 16 + lh;
  const int c0 = wave * 64;            // 64-col group

  const float* xrow = x + (size_t)m * DD + 2 * ksel;
  const float* wr0  = w + (size_t)(c0 + 0 * 16 + lh) * DD + 2 * ksel;
  const float* wr1  = w + (size_t)(c0 + 1 * 16 + lh) * DD + 2 * ksel;
  const float* wr2  = w + (size_t)(c0 + 2 * 16 + lh) * DD + 2 * ksel;
  const float* wr3  = w + (size_t)(c0 + 3 * 16 + lh) * DD + 2 * ksel;

  v8f acc0 = {}, acc1 = {}, acc2 = {}, acc3 = {};

#pragma unroll 4
  for (int kk = 0; kk < DD / 4; ++kk) {
    const int o = 4 * kk;
    v2f a = ld2(xrow + o);
    acc0 = wmma4(a, ld2(wr0 + o), acc0);
    acc1 = wmma4(a, ld2(wr1 + o), acc1);
    acc2 = wmma4(a, ld2(wr2 + o), acc2);
    acc3 = wmma4(a, ld2(wr3 + o), acc3);
  }

  // Epilogue: C/D layout -> VGPR j holds row (mt*16 + j + 8*ksel), col = lh
  const int rowbase = mt * 16 + 8 * ksel;
  v8f accs[4] = {acc0, acc1, acc2, acc3};
#pragma unroll
  for (int t = 0; t < 4; ++t) {
    const int col = c0 + t * 16 + lh;
    const float bv_ = bias[col];
#pragma unroll
    for (int j = 0; j < 8; ++j)
      out[(size_t)(rowbase + j) * DD + col] = accs[t][j] + bv_;
  }
}

// ---------------------------------------------------------------------------
// Kernel 2: spinor flash attention.  grid=(N/64, H, B), block=128 (4 waves).
// All 4 waves share one (b,h); K/V 16x32 tiles are staged in LDS once per
// block via double-buffered gfx1250 async copies (ASYNCcnt), overlapping the
// next tile's DMA with the current tile's 24 WMMAs.
// Scores are bounded in [0, 0.177] after scaling -> plain sum-of-exp softmax.
// re uses K directly; im uses the per-lane pair-rotated K (free swap+negate).
// P (exp weights) goes C-layout -> LDS (padded) -> A-layout for the PV WMMAs.
// ---------------------------------------------------------------------------
__global__ __launch_bounds__(128)
void spinor_attn_kernel(const float* __restrict__ qb,
                        const float* __restrict__ kb,
                        const float* __restrict__ vb,
                        float* __restrict__ ob)
{
  extern __shared__ float smem[];

  const int tid  = threadIdx.x;
  const int wave = tid >> 5;
  const int lane = tid & 31;
  const int lh   = lane & 15;
  const int ksel = lane >> 4;

  const int b  = blockIdx.z;
  const int h  = blockIdx.y;
  const int qt = blockIdx.x * 4 + wave;   // query tile 0..127
  const int n0 = qt * 16;

  // Cooperative tile loader: thread t moves one 16B chunk of K and of V.
  const int lr = tid >> 3;                // tile row 0..15
  const int lc = (tid & 7) * 4;           // tile col 0,4,...,28
  const size_t gtile = (size_t)(b * NN) * DD + h * HD + lc;
  const unsigned lds_k = (unsigned)((KT_OFF + lr * TILE_STRIDE + lc) * 4);
  const unsigned lds_v = (unsigned)((VT_OFF + lr * TILE_STRIDE + lc) * 4);

  // Prologue: DMA tile 0 into buffer 0.
  async_copy_b128(lds_k, kb + gtile + (size_t)lr * DD);
  async_copy_b128(lds_v, vb + gtile + (size_t)lr * DD);

  // Preload 8 Q A-fragments (d = 0..31 in steps of 4)
  const float* qp = qb + (size_t)(b * NN + n0 + lh) * DD + h * HD + 2 * ksel;
  v2f qa[8];
#pragma unroll
  for (int s = 0; s < 8; ++s) qa[s] = ld2(qp + 4 * s);

  v8f o0 = {}, o1 = {};     // PV accumulators, d in [0,16) and [16,32)
  v8f lsum = {};            // per-lane partial sum of exp-weights

  float* myp = smem + PB_OFF + wave * 320;   // 16x20 padded P tile
  const unsigned bufstep = (unsigned)(16 * TILE_STRIDE);

  int buf = 0;
  for (int mt = 0; mt < NN / 16; ++mt) {
    // My DMA landed; after the barrier everyone's has, and every wave is done
    // reading the buffer we are about to refill (last touched at mt-1).
    asm volatile("s_wait_asynccnt 0" ::: "memory");
    __syncthreads();

    if (mt + 1 < NN / 16) {
      const size_t gnext = gtile + (size_t)((mt + 1) * 16 + lr) * DD;
      const unsigned lo = (buf ^ 1) * bufstep * 4;
      async_copy_b128(lds_k + lo, kb + gnext);
      async_copy_b128(lds_v + lo, vb + gnext);
    }

    const float* ks = smem + KT_OFF + buf * bufstep;
    const float* vs = smem + VT_OFF + buf * bufstep;

    // ---- scores: re = q.k , im = q.rot(k) ----
    v8f re = {}, im = {};
    const float* krow = ks + lh * TILE_STRIDE + 2 * ksel;
#pragma unroll
    for (int s = 0; s < 8; ++s) {
      v2f kf = ld2(krow + 4 * s);
      re = wmma4(qa[s], kf, re);
      v2f kr; kr[0] = -kf[1]; kr[1] = kf[0];   // pair rotation (i * conj trick)
      im = wmma4(qa[s], kr, im);
    }

    // ---- weight + exp (closed form: cos(atan2(im,re)) = re*rsqrt(re^2+im^2))
    v8f p;
#pragma unroll
    for (int j = 0; j < 8; ++j) {
      float r = re[j], i = im[j];
      float den = r * r + i * i;
      float c = (den > 0.0f) ? r * __frsqrt_rn(den) : 1.0f;
      float wv = 0.5f * (1.0f + c) * INV_SCALE;
      float e = __expf(wv);
      p[j] = e;
      lsum[j] += e;
    }

    // ---- transpose P through LDS: C-layout store, A-layout load ----
#pragma unroll
    for (int j = 0; j < 8; ++j)
      myp[(j + 8 * ksel) * 20 + lh] = p[j];
    asm volatile("s_wait_dscnt 0" ::: "memory");   // wave-local RAW turn-around

    // ---- PV: out += P(16x16) @ V(16x32) ----
#pragma unroll
    for (int kk = 0; kk < 4; ++kk) {
      v2f pa = ld2(myp + lh * 20 + 4 * kk + 2 * ksel);
      const float* vrow = vs + (4 * kk + 2 * ksel) * TILE_STRIDE;
      v2f b0, b1;
      b0[0] = vrow[lh];                    b0[1] = vrow[TILE_STRIDE + lh];
      b1[0] = vrow[16 + lh];               b1[1] = vrow[TILE_STRIDE + 16 + lh];
      o0 = wmma4(pa, b0, o0);
      o1 = wmma4(pa, b1, o1);
    }

    buf ^= 1;
  }

  // ---- softmax normalization: reduce exp-sums over the 16 columns ----
#pragma unroll
  for (int j = 0; j < 8; ++j) {
    float s = lsum[j];
    s += __shfl_xor(s, 1);
    s += __shfl_xor(s, 2);
    s += __shfl_xor(s, 4);
    s += __shfl_xor(s, 8);                 // stays within each 16-lane half
    const float inv = __frcp_rn(s);
    const size_t ro = (size_t)(b * NN + n0 + j + 8 * ksel) * DD + h * HD;
    ob[ro + lh]      = o0[j] * inv;
    ob[ro + 16 + lh] = o1[j] * inv;
  }
}

// ---------------------------------------------------------------------------
// Kernel 3: gate MLP (Linear->SiLU->Linear->Sigmoid) + final multiply.
// grid=4096/4, block=128 (one wave per row).  Lane l owns gate hidden units
// j=l and j=l+32; the row is staged in LDS and broadcast-read.
// ---------------------------------------------------------------------------
__global__ __launch_bounds__(128)
void gate_kernel(const float* __restrict__ ob,
                 const float* __restrict__ wg1, const float* __restrict__ bg1,
                 const float* __restrict__ wg2, const float* __restrict__ bg2,
                 float* __restrict__ out)
{
  __shared__ float rowbuf[4][DD];
  const int wave = threadIdx.x >> 5;
  const int lane = threadIdx.x & 31;
  const int n = blockIdx.x * 4 + wave;

  const float* orow = ob + (size_t)n * DD;
  float* row = rowbuf[wave];
#pragma unroll
  for (int t = 0; t < DD / 32; ++t) row[lane + 32 * t] = orow[lane + 32 * t];
  asm volatile("s_wait_dscnt 0" ::: "memory");

  float acc0 = bg1[lane];
  float acc1 = bg1[lane + 32];
  const float* w0 = wg1 + (size_t)lane * DD;
  const float* w1 = wg1 + (size_t)(lane + 32) * DD;
#pragma unroll 4
  for (int k = 0; k < DD; k += 4) {
    float4 xv = *reinterpret_cast<const float4*>(&row[k]);
    float4 a0 = *reinterpret_cast<const float4*>(w0 + k);
    float4 a1 = *reinterpret_cast<const float4*>(w1 + k);
    acc0 += xv.x * a0.x + xv.y * a0.y + xv.z * a0.z + xv.w * a0.w;
    acc1 += xv.x * a1.x + xv.y * a1.y + xv.z * a1.z + xv.w * a1.w;
  }
  const float h0 = acc0 * sigmoidf_(acc0);   // SiLU
  const float h1 = acc1 * sigmoidf_(acc1);

  float g = h0 * wg2[lane] + h1 * wg2[lane + 32];
  g += __shfl_xor(g, 1);
  g += __shfl_xor(g, 2);
  g += __shfl_xor(g, 4);
  g += __shfl_xor(g, 8);
  g += __shfl_xor(g, 16);
  g = sigmoidf_(g + bg2[0]);

#pragma unroll
  for (int t = 0; t < DD / 32; ++t)
    out[(size_t)n * DD + lane + 32 * t] = row[lane + 32 * t] * g;
}

// ---------------------------------------------------------------------------
extern "C" void kernel_launch(void* const* d_in, const int* in_sizes, int n_in,
                              void* d_out, int out_size, void* d_ws,
                              size_t ws_size, hipStream_t stream)
{
  const float* x   = (const float*)d_in[0];
  const float* wq  = (const float*)d_in[1];
  const float* bq  = (const float*)d_in[2];
  const float* wk  = (const float*)d_in[3];
  const float* bk  = (const float*)d_in[4];
  const float* wv  = (const float*)d_in[5];
  const float* bv  = (const float*)d_in[6];
  const float* wg1 = (const float*)d_in[7];
  const float* bg1 = (const float*)d_in[8];
  const float* wg2 = (const float*)d_in[9];
  const float* bg2 = (const float*)d_in[10];
  float* out = (float*)d_out;

  // Workspace: q | k | v | attn_out   (4 x 4 MB = 16 MB)
  float* qb = (float*)d_ws;
  float* kb = qb + (size_t)MROWS * DD;
  float* vb = kb + (size_t)MROWS * DD;
  float* ob = vb + (size_t)MROWS * DD;

  {
    dim3 grid(MROWS / 16, 3);
    qkv_proj_kernel<<<grid, 128, 0, stream>>>(x, wq, bq, wk, bk, wv, bv,
                                              qb, kb, vb);
  }
  {
    dim3 grid(NN / 64, HH, BB);
    spinor_attn_kernel<<<grid, 128, SMEM_BYTES, stream>>>(qb, kb, vb, ob);
  }
  {
    dim3 grid(MROWS / 4);
    gate_kernel<<<grid, 128, 0, stream>>>(ob, wg1, bg1, wg2, bg2, out);
  }
}